// NRPreprocessing_81363860455910
// MI455X (gfx1250) — compile-verified
//
#include <hip/hip_runtime.h>
#include <hip/hip_bf16.h>

// Problem constants (match reference setup_inputs)
constexpr int R  = 12;    // NUM_RES_PER_PRB
constexpr int T  = 14;    // OFDM symbols
constexpr int F  = 3276;  // subcarriers (= num_prbs * 12)
constexpr int CH = 8;     // channels

typedef float v4f __attribute__((ext_vector_type(4)));
typedef float v2f __attribute__((ext_vector_type(2)));

// h_out[b,0,f,t,c] = val_b[ nn[t][f%12] ], broadcast over c (8 channels).
// val_b[k] = 0.5*(h_hat[b,0,0,6k] + h_hat[b,0,0,6k+1]).
// nn replicates the reference's grid-flatten/reshape quirk exactly.
__global__ void __launch_bounds__(256) nrpre_hout_kernel(
    const float* __restrict__ h_hat,
    const int*   __restrict__ sym_pos,   // dmrs_ofdm_pos, 2 entries
    const int*   __restrict__ sc_pos,    // dmrs_subcarrier_pos, 6 entries
    float*       __restrict__ h_out)
{
    __shared__ float raw[R * 2];    // staged h_hat pairs (async DMA target)
    __shared__ float val[R];
    __shared__ int   nn_s[R * T];   // indexed by g = t'*12 + r'
    __shared__ float tval[R * T];   // tval[t*12 + r] = val[nn[t][r]]

    const int tid = threadIdx.x;
    const int b   = blockIdx.y;

    const float* hb = h_hat + (size_t)b * (CH * F);  // channel 0 slice of batch b

    // Stage the 12 FOCC pairs straight into LDS via the gfx1250 async
    // global->LDS engine (no VGPR round-trip, tracked by ASYNCcnt).
    if (tid < R) {
        const unsigned long long gaddr = (unsigned long long)(const void*)(hb + 6 * tid);
        const unsigned lds_off = (unsigned)(uintptr_t)(&raw[2 * tid]);
        asm volatile("global_load_async_to_lds_b64 %0, %1, off"
                     :: "v"(lds_off), "v"(gaddr) : "memory");
    }
    asm volatile("s_wait_asynccnt 0x0" ::: "memory");

    if (tid < R * T) {
        // grid flat index g -> (r = g/14, t = g%14); reshaped as nn[t'][r'] with g = t'*12+r'
        const int g = tid;
        const int r = g / T;
        const int t = g % T;
        int best = 0, bestd = 1 << 30;
#pragma unroll
        for (int p = 0; p < 12; ++p) {           // p = i*2 + j (6 subcarriers x 2 symbols)
            const int i = p >> 1, j = p & 1;
            const int d = __builtin_abs(r - sc_pos[i]) + __builtin_abs(t - sym_pos[j]);
            if (d < bestd) { bestd = d; best = p; }   // first-min tie-break like jnp.argmin
        }
        nn_s[g] = best;
    }
    __syncthreads();
    if (tid < R) val[tid] = 0.5f * (raw[2 * tid] + raw[2 * tid + 1]);
    __syncthreads();
    if (tid < R * T) tval[tid] = val[nn_s[tid]];
    __syncthreads();

    // Block owns f in [f0, f0+256): a contiguous slab of fcount*112 floats.
    // Enumerate its float4 chunks q so consecutive lanes hit consecutive 16B
    // addresses -> each wave32 store instruction writes 512B contiguous.
    const int f0     = blockIdx.x * 256;
    const int fcount = min(256, F - f0);
    if (fcount <= 0) return;
    const int nq = fcount * (T * CH / 4);            // float4 chunks in slab (28 per f)
    v4f* bp = (v4f*)(h_out + ((size_t)b * F + f0) * (size_t)(T * CH));

    for (int q = tid; q < nq; q += 256) {
        const int f_local = q / 28;                  // chunk -> (f, t, half)
        const int rem     = q - f_local * 28;
        const int t       = rem >> 1;
        const float v  = tval[t * R + ((f0 + f_local) % R)];
        const v4f vv   = {v, v, v, v};
        __builtin_nontemporal_store(vv, bp + q);     // global_store_b128 ... th:TH_STORE_NT
    }
}

// pe[0,f,t,c] = norm_pe[f%12, t, c] : constant table, demeaned + /std(ddof=1).
__global__ void __launch_bounds__(256) nrpre_pe_kernel(
    const int* __restrict__ sym_pos,
    const int* __restrict__ sc_pos,
    float*     __restrict__ pe_out)
{
    __shared__ float p0[R * T], p1[R * T];   // indexed by g = t'*12 + r'
    __shared__ float tab[R * T * 2];         // [(f%12)*14 + t]*2 + c
    __shared__ float stats[4];               // mean0, mean1, sd0, sd1

    const int tid = threadIdx.x;
    if (tid < R * T) {
        const int g = tid;
        const int r = g / T;                 // grid coords (flatten order g = r*14+t)
        const int t = g % T;
        int m0 = 1 << 30, m1 = 1 << 30;
#pragma unroll
        for (int i = 0; i < 6; ++i) m0 = min(m0, __builtin_abs(r - sc_pos[i]));
#pragma unroll
        for (int j = 0; j < 2; ++j) m1 = min(m1, __builtin_abs(t - sym_pos[j]));
        p0[g] = (float)m0;
        p1[g] = (float)m1;
    }
    __syncthreads();
    if (tid == 0) {
        float s0 = 0.f, s1 = 0.f;
        for (int g = 0; g < R * T; ++g) { s0 += p0[g]; s1 += p1[g]; }
        stats[0] = s0 / 168.f;
        stats[1] = s1 / 168.f;
    }
    __syncthreads();
    if (tid < R * T) { p0[tid] -= stats[0]; p1[tid] -= stats[1]; }
    __syncthreads();
    if (tid == 0) {
        // jnp.std on the demeaned array: subtract its own (near-zero) mean, ddof=1
        float s0 = 0.f, s1 = 0.f;
        for (int g = 0; g < R * T; ++g) { s0 += p0[g]; s1 += p1[g]; }
        const float m0 = s0 / 168.f, m1 = s1 / 168.f;
        float q0 = 0.f, q1 = 0.f;
        for (int g = 0; g < R * T; ++g) {
            const float d0 = p0[g] - m0, d1 = p1[g] - m1;
            q0 += d0 * d0; q1 += d1 * d1;
        }
        stats[2] = sqrtf(q0 / 167.f);
        stats[3] = sqrtf(q1 / 167.f);
    }
    __syncthreads();
    if (tid < R * T) {
        const int g  = tid;                  // g = t'*12 + r'
        const int tp = g / R, rp = g % R;
        const float sd0 = stats[2], sd1 = stats[3];
        tab[(rp * T + tp) * 2 + 0] = (sd0 > 0.f) ? p0[g] / sd0 : p0[g];
        tab[(rp * T + tp) * 2 + 1] = (sd1 > 0.f) ? p1[g] / sd1 : p1[g];
    }
    __syncthreads();

    const int total  = F * T;                // 45,864 (f,t) pairs, 2 floats each
    const int stride = gridDim.x * blockDim.x;
    for (int idx = blockIdx.x * blockDim.x + tid; idx < total; idx += stride) {
        const int f = idx / T, t = idx - f * T;
        const int base = ((f % R) * T + t) * 2;
        const v2f vv = { tab[base], tab[base + 1] };
        __builtin_nontemporal_store(vv, (v2f*)pe_out + idx);
    }
}

extern "C" void kernel_launch(void* const* d_in, const int* in_sizes, int n_in,
                              void* d_out, int out_size, void* d_ws, size_t ws_size,
                              hipStream_t stream) {
    // setup_inputs order: y (unused), h_hat, dmrs_ofdm_pos, dmrs_subcarrier_pos
    const float* h_hat = (const float*)d_in[1];
    const int*   sym   = (const int*)d_in[2];
    const int*   sc    = (const int*)d_in[3];
    float*       out   = (float*)d_out;

    const int bs = in_sizes[1] / (CH * F);   // 128

    dim3 grid((F + 255) / 256, bs);          // 13 x 128 blocks
    nrpre_hout_kernel<<<grid, 256, 0, stream>>>(h_hat, sym, sc, out);

    float* pe_out = out + (size_t)bs * F * (size_t)(T * CH);  // h_out is 46,964,736 floats
    nrpre_pe_kernel<<<dim3(30), 256, 0, stream>>>(sym, sc, pe_out);
}